// Response_score_44401371906485
// MI455X (gfx1250) — compile-verified
//
#include <hip/hip_runtime.h>
#include <math.h>
#include <stdint.h>

// ---------------------------------------------------------------------------
// Additive attention scores (B=4, Q=K=512, D=512, H=256):
//   qp = queries(2048,512) @ Wq(512,256)      [fp32 WMMA]
//   kp = keys(2048,512)    @ Wk(512,256)      [fp32 WMMA]
//   out[b,q,k] = sum_h tanh(qp[b,q,h] + kp[b,k,h]) * wv[h]   [v_tanh_f32]
// ---------------------------------------------------------------------------

typedef float v2f __attribute__((ext_vector_type(2)));
typedef float v8f __attribute__((ext_vector_type(8)));

#define BQ   2048      // B*Q rows (also B*K)
#define DIN  512       // projection input dim
#define HID  256       // hidden dim
#define LDSS 260       // padded LDS row stride (floats): 260*4 = 1040 B, 16B-aligned

static __device__ __forceinline__ float fast_tanh(float x) {
#if __has_builtin(__builtin_amdgcn_tanhf)
    return __builtin_amdgcn_tanhf(x);   // native v_tanh_f32 (TRANS pipe, gfx1250)
#else
    return tanhf(x);
#endif
}

// Async global->LDS b128 copy (per-lane). dst must be a generic pointer to
// __shared__; its low 32 bits are the LDS byte offset (LDS aperture layout).
static __device__ __forceinline__ void async_copy_b128(void* lds_dst, const void* gsrc) {
    uint32_t loff = (uint32_t)(uintptr_t)lds_dst;
    uint64_t gaddr = (uint64_t)(uintptr_t)gsrc;
    asm volatile("global_load_async_to_lds_b128 %0, %1, off"
                 :: "v"(loff), "v"(gaddr) : "memory");
}

static __device__ __forceinline__ void async_wait0() {
#if __has_builtin(__builtin_amdgcn_s_wait_asynccnt)
    __builtin_amdgcn_s_wait_asynccnt(0);
#else
    asm volatile("s_wait_asynccnt 0x0" ::: "memory");
#endif
}

// ---------------------------------------------------------------------------
// Projection GEMM: Y(2048,256) = X(2048,512) @ W(512,256), fp32 WMMA 16x16x4.
// One wave per 16x16 output tile; 8 waves per block; 2048 tiles total.
//
// fp32 A (16x4, MxK): lanes 0-15 hold M=lane, VGPR0=K+0, VGPR1=K+1;
//                     lanes 16-31 hold M=lane-16, VGPR0=K+2, VGPR1=K+3.
// fp32 B (4x16, KxN) mirrored; fp32 C/D: VGPR j -> M=j (+8 for lanes 16-31).
// ---------------------------------------------------------------------------
__global__ void __launch_bounds__(256)
proj_wmma_f32(const float* __restrict__ X,
              const float* __restrict__ W,
              float* __restrict__ Y)
{
    const int lane = threadIdx.x & 31;
    const int wave = threadIdx.x >> 5;
    const int tile = blockIdx.x * 8 + wave;       // 0 .. 2047
    const int tn   = tile & 15;                   // 16 N-tiles
    const int tm   = tile >> 4;                   // 128 M-tiles
    const int m0   = tm * 16;
    const int n0   = tn * 16;
    const int half = lane >> 4;                   // 0: K+0/1, 1: K+2/3
    const int l    = lane & 15;

    v8f acc = {};
    const float* xrow = X + (size_t)(m0 + l) * DIN + 2 * half;
    const float* wcol = W + (size_t)(2 * half) * HID + n0 + l;

    #pragma unroll 4
    for (int k = 0; k < DIN; k += 4) {
        v2f a, b;
        a.x = xrow[k + 0];                        // X[m, k+2h+0]
        a.y = xrow[k + 1];                        // X[m, k+2h+1]
        b.x = wcol[(size_t)k * HID];              // W[k+2h+0, n]
        b.y = wcol[(size_t)(k + 1) * HID];        // W[k+2h+1, n]
        acc = __builtin_amdgcn_wmma_f32_16x16x4_f32(
                  /*neg_a=*/false, a, /*neg_b=*/false, b,
                  /*c_mod=*/(short)0, acc,
                  /*reuse_a=*/false, /*reuse_b=*/false);
    }

    float* yout = Y + (size_t)(m0 + 8 * half) * HID + n0 + l;
    #pragma unroll
    for (int j = 0; j < 8; ++j)
        yout[(size_t)j * HID] = acc[j];
}

// ---------------------------------------------------------------------------
// Score kernel: block = 128 threads (4 waves), tile = 16 q-rows x 32 k-cols
// for one batch. LDS tiles staged via global_load_async_to_lds_b128.
// Wave w owns q-rows {4w..4w+3}; lanes own the 32 k-cols; h-loop outermost so
// each per-lane kv float4 is reused across 4 accumulators (TRANS-bound stage:
// 1 v_tanh + 2 VALU + 0.375 ds_b128 per output-element-hidden).
// ---------------------------------------------------------------------------
__global__ void __launch_bounds__(128)
score_tanh(const float* __restrict__ qp,   // (B*Q, 256)
           const float* __restrict__ kp,   // (B*K, 256)
           const float* __restrict__ wv,   // (256)
           float* __restrict__ out)        // (B, Q, K)
{
    __shared__ float s_kp[32 * LDSS];      // 33,280 B
    __shared__ float s_qp[16 * LDSS];      // 16,640 B
    __shared__ float s_wv[HID];            //  1,024 B

    const int b  = blockIdx.z;
    const int q0 = blockIdx.y * 16;
    const int k0 = blockIdx.x * 32;

    // --- stage kp tile: 32 rows x 64 float4 (async) ---
    for (int idx = threadIdx.x; idx < 32 * 64; idx += 128) {
        const int r = idx >> 6, c = (idx & 63) * 4;
        async_copy_b128(&s_kp[r * LDSS + c],
                        &kp[((size_t)(b * 512 + k0 + r)) * HID + c]);
    }
    // --- stage qp tile: 16 rows x 64 float4 (async) ---
    for (int idx = threadIdx.x; idx < 16 * 64; idx += 128) {
        const int r = idx >> 6, c = (idx & 63) * 4;
        async_copy_b128(&s_qp[r * LDSS + c],
                        &qp[((size_t)(b * 512 + q0 + r)) * HID + c]);
    }
    // --- stage wv: 64 float4 (async) ---
    if (threadIdx.x < 64)
        async_copy_b128(&s_wv[threadIdx.x * 4], &wv[threadIdx.x * 4]);

    async_wait0();
    __syncthreads();

    const int lane = threadIdx.x & 31;
    const int wave = threadIdx.x >> 5;
    const float* krow = &s_kp[lane * LDSS];
    const float* qr0  = &s_qp[(wave * 4 + 0) * LDSS];
    const float* qr1  = &s_qp[(wave * 4 + 1) * LDSS];
    const float* qr2  = &s_qp[(wave * 4 + 2) * LDSS];
    const float* qr3  = &s_qp[(wave * 4 + 3) * LDSS];

    float acc0 = 0.0f, acc1 = 0.0f, acc2 = 0.0f, acc3 = 0.0f;

    #pragma unroll 2
    for (int h = 0; h < HID; h += 4) {
        const float4 kv = *(const float4*)&krow[h];   // per-lane ds_load_b128
        const float4 wf = *(const float4*)&s_wv[h];   // broadcast
        const float4 a0 = *(const float4*)&qr0[h];    // broadcast
        const float4 a1 = *(const float4*)&qr1[h];
        const float4 a2 = *(const float4*)&qr2[h];
        const float4 a3 = *(const float4*)&qr3[h];

        acc0 = fmaf(fast_tanh(a0.x + kv.x), wf.x, acc0);
        acc1 = fmaf(fast_tanh(a1.x + kv.x), wf.x, acc1);
        acc2 = fmaf(fast_tanh(a2.x + kv.x), wf.x, acc2);
        acc3 = fmaf(fast_tanh(a3.x + kv.x), wf.x, acc3);

        acc0 = fmaf(fast_tanh(a0.y + kv.y), wf.y, acc0);
        acc1 = fmaf(fast_tanh(a1.y + kv.y), wf.y, acc1);
        acc2 = fmaf(fast_tanh(a2.y + kv.y), wf.y, acc2);
        acc3 = fmaf(fast_tanh(a3.y + kv.y), wf.y, acc3);

        acc0 = fmaf(fast_tanh(a0.z + kv.z), wf.z, acc0);
        acc1 = fmaf(fast_tanh(a1.z + kv.z), wf.z, acc1);
        acc2 = fmaf(fast_tanh(a2.z + kv.z), wf.z, acc2);
        acc3 = fmaf(fast_tanh(a3.z + kv.z), wf.z, acc3);

        acc0 = fmaf(fast_tanh(a0.w + kv.w), wf.w, acc0);
        acc1 = fmaf(fast_tanh(a1.w + kv.w), wf.w, acc1);
        acc2 = fmaf(fast_tanh(a2.w + kv.w), wf.w, acc2);
        acc3 = fmaf(fast_tanh(a3.w + kv.w), wf.w, acc3);
    }

    const size_t obase = ((size_t)(b * 512 + q0 + wave * 4)) * 512 + k0 + lane;
    out[obase + 0 * 512] = acc0;
    out[obase + 1 * 512] = acc1;
    out[obase + 2 * 512] = acc2;
    out[obase + 3 * 512] = acc3;
}

// ---------------------------------------------------------------------------
extern "C" void kernel_launch(void* const* d_in, const int* in_sizes, int n_in,
                              void* d_out, int out_size, void* d_ws, size_t ws_size,
                              hipStream_t stream) {
    const float* queries = (const float*)d_in[0];  // (4,512,1,512)  = (2048,512)
    const float* keys    = (const float*)d_in[1];  // (4,1,512,512)  = (2048,512)
    const float* Wq      = (const float*)d_in[2];  // (512,256)
    const float* Wk      = (const float*)d_in[3];  // (512,256)
    const float* wv      = (const float*)d_in[4];  // (256)
    float* out = (float*)d_out;                    // (4,512,512)

    float* qp = (float*)d_ws;                      // 2048*256 floats = 2 MB
    float* kp = qp + (size_t)BQ * HID;             // 2048*256 floats = 2 MB

    // Projections: 2048 tiles of 16x16, 8 waves/block -> 256 blocks.
    proj_wmma_f32<<<dim3(256), dim3(256), 0, stream>>>(queries, Wq, qp);
    proj_wmma_f32<<<dim3(256), dim3(256), 0, stream>>>(keys,    Wk, kp);

    // Scores: grid (k-tiles=16, q-tiles=32, batch=4), 128 threads/block.
    score_tanh<<<dim3(16, 32, 4), dim3(128), 0, stream>>>(qp, kp, wv, out);
}